// PINNExtension_39822936768956
// MI455X (gfx1250) — compile-verified
//
#include <hip/hip_runtime.h>
#include <cstdint>

// ---------------------------------------------------------------------------
// Fused depthwise FD(5-tap) + Gaussian(9-tap) valid conv == one 13-tap conv.
//   * HBM-bound: 128MB in + 128MB out @ 23.3 TB/s  =>  ~11us floor.
//   * MACs routed to the CDNA5 matrix pipe: banded-Toeplitz GEMM
//       D[16x16] = A[16x28] x B[28x16]   (7x V_WMMA_F32_16X16X4_F32)
//   * Staging: GLOBAL_LOAD_ASYNC_TO_LDS_B128 (ASYNCcnt) — direct mem->LDS,
//     no VGPR bounce; falls back to float4 copies if builtin is absent.
// ---------------------------------------------------------------------------

typedef float v2f __attribute__((ext_vector_type(2)));
typedef float v8f __attribute__((ext_vector_type(8)));
typedef int   b128_t __attribute__((vector_size(16)));   // builtin's pointee type

#define ROWS   32      // B*C = 8*4 independent 1-D streams
#define HALO   12      // 13-tap valid conv halo
#define TILE   128     // output columns per block (8 waves x 16)
#define LROWS  16      // rows per block tile
#define LSTR   140     // LDS row stride in floats (128 + 12, conflict-free)
#define WPADN  43      // zero-padded composed kernel: indices [-15..27] + 15

#if defined(__has_builtin)
#if __has_builtin(__builtin_amdgcn_global_load_async_to_lds_b128)
#define HAS_ASYNC_LDS 1
#endif
#endif

// Flat->AS cast helpers (flat LDS address low 32 bits == LDS byte offset).
#define AS1V(p) ((__attribute__((address_space(1))) b128_t*)(uintptr_t)(p))
#define AS3V(p) ((__attribute__((address_space(3))) b128_t*)(uint32_t)(uintptr_t)(p))

__global__ __launch_bounds__(256)
void fd_gauss_wmma(const float* __restrict__ x,
                   const float* __restrict__ fd,
                   const float* __restrict__ gs,
                   float* __restrict__ out,
                   int T, int OUT)
{
    __shared__ float lds[LROWS * LSTR + 48];   // tile + padded kernel
    float* wpad = &lds[LROWS * LSTR];

    const int tid     = threadIdx.x;
    const int rowBase = blockIdx.y * LROWS;
    const int colBase = blockIdx.x * TILE;

    // ---- stage 16 x 140 input tile into LDS ----
    for (int i = tid; i < LROWS * 35; i += 256) {
        const int r  = i / 35;
        const int c  = (i - r * 35) * 4;
        const int gc = colBase + c;
        const float* src = x + (size_t)(rowBase + r) * (size_t)T;
        if (gc + 3 < T) {
#ifdef HAS_ASYNC_LDS
            // direct memory->LDS, tracked on ASYNCcnt (no VGPR data round trip)
            __builtin_amdgcn_global_load_async_to_lds_b128(
                AS1V(src + gc), AS3V(&lds[r * LSTR + c]), 0, 0);
#else
            *(float4*)(&lds[r * LSTR + c]) = *(const float4*)(src + gc);
#endif
        } else {                                // tail block: clamp reads
            const int tmax = T - 1;
            float4 v;
            v.x = src[gc     < T ? gc     : tmax];
            v.y = src[gc + 1 < T ? gc + 1 : tmax];
            v.z = src[gc + 2 < T ? gc + 2 : tmax];
            v.w = src[gc + 3 < T ? gc + 3 : tmax];
            *(float4*)(&lds[r * LSTR + c]) = v;
        }
    }

    // ---- compose 13-tap kernel w = fd (*) gauss into zero-padded wpad[43] ----
    if (tid < WPADN) {
        float val = 0.f;
        const int j = tid - 15;                 // tap index
        if (j >= 0 && j <= 12) {
#pragma unroll
            for (int i5 = 0; i5 < 5; ++i5) {
                const int gi = j - i5;
                if (gi >= 0 && gi <= 8) val += fd[i5] * gs[gi];
            }
        }
        wpad[tid] = val;
    }

    // ---- L2 warm-up for the neighboring column tile (global_prefetch_b8) ----
    {
        const int pr = tid & 15;
        const int pc = colBase + TILE + ((tid >> 4) * 16);
        if (pc < T)
            __builtin_prefetch(x + (size_t)(rowBase + pr) * (size_t)T + pc, 0, 0);
    }

#ifdef HAS_ASYNC_LDS
#if __has_builtin(__builtin_amdgcn_s_wait_asynccnt)
    __builtin_amdgcn_s_wait_asynccnt(0);        // my async LDS writes landed
#else
    asm volatile("s_wait_asynccnt 0x0" ::: "memory");
#endif
#endif
    __syncthreads();                            // all waves' writes visible

    const int wave   = tid >> 5;
    const int lane   = tid & 31;
    const int laneM  = lane & 15;               // M for A / N for B,D
    const int laneHi = lane >> 4;               // selects K pair / M+8 half

    // ---- banded B fragments: B[kk][n] = w[kk - n], kk = 4c + 2*laneHi + {0,1}
    v2f bf[7];
#pragma unroll
    for (int c = 0; c < 7; ++c) {
        const int bi = 4 * c + 2 * laneHi - laneM + 15;   // in [0, 42]
        bf[c].x = wpad[bi];
        bf[c].y = wpad[bi + 1];
    }

    v8f acc = {};

#if __has_builtin(__builtin_amdgcn_wmma_f32_16x16x4_f32)
#pragma unroll
    for (int c = 0; c < 7; ++c) {
        // A[m][kk]: lane m holds kk = 4c + 2*laneHi + {0,1}; even offset -> 8B aligned
        const v2f a = *(const v2f*)(&lds[laneM * LSTR + wave * 16 + 4 * c + 2 * laneHi]);
        acc = __builtin_amdgcn_wmma_f32_16x16x4_f32(
            false, a, false, bf[c], (short)0, acc, false, false);
    }
#else
    // VALU fallback producing the same D-layout (lane=N, vgpr v -> M = v + 8*laneHi)
#pragma unroll
    for (int v = 0; v < 8; ++v) {
        const int m = v + 8 * laneHi;
        float s = 0.f;
#pragma unroll
        for (int k = 0; k < 13; ++k)
            s += wpad[k + 15] * lds[m * LSTR + wave * 16 + laneM + k];
        acc[v] = s;
    }
#endif

    // ---- store D: vgpr v -> row (v + 8*laneHi), col = laneM; mask tail cols ----
    const int gc = colBase + wave * 16 + laneM;
    if (gc < OUT) {
#pragma unroll
        for (int v = 0; v < 8; ++v) {
            const int m = v + 8 * laneHi;
            out[(size_t)(rowBase + m) * (size_t)OUT + gc] = acc[v];
        }
    }
}

extern "C" void kernel_launch(void* const* d_in, const int* in_sizes, int n_in,
                              void* d_out, int out_size, void* d_ws, size_t ws_size,
                              hipStream_t stream)
{
    const float* x  = (const float*)d_in[0];   // [8,4,T] fp32
    const float* fd = (const float*)d_in[1];   // 5 taps
    const float* gs = (const float*)d_in[2];   // 9 taps
    float*       o  = (float*)d_out;           // [8,4,T-12] fp32

    const int T   = in_sizes[0] / ROWS;        // 1048576
    const int OUT = T - HALO;                  // 1048564

    dim3 grid((OUT + TILE - 1) / TILE, ROWS / LROWS);
    fd_gauss_wmma<<<grid, 256, 0, stream>>>(x, fd, gs, o, T, OUT);
}